// Net_59674275610883
// MI455X (gfx1250) — compile-verified
//
#include <hip/hip_runtime.h>

typedef float v2f __attribute__((ext_vector_type(2)));
typedef float v8f __attribute__((ext_vector_type(8)));

#define B_SZ   4096
#define D_IN   784
#define D_H    2048
#define D_OUT  10
#define STEPS  25
#define BETA   0.95f

// ---------------------------------------------------------------------------
// Kernel 1: cur1 = x @ W1^T + b1   via V_WMMA_F32_16X16X4_F32
// Block: 256 thr (8 waves in 2x4). Block tile 128x128, wave tile 64x32
// (4x2 WMMA 16x16 tiles). K staged through LDS in chunks of 16 (784 = 49*16).
// LDS rows padded to 18 floats -> conflict-free 16-row strided frag reads.
// ---------------------------------------------------------------------------
__global__ __launch_bounds__(256) void gemm1_wmma(
    const float* __restrict__ X, const float* __restrict__ W1,
    const float* __restrict__ b1, float* __restrict__ C)
{
  __shared__ float XS[128][18];
  __shared__ float WS[128][18];

  const int tid  = threadIdx.x;
  const int lane = tid & 31;
  const int wid  = tid >> 5;
  const int wm   = wid >> 2;      // 0..1  (M direction)
  const int wn   = wid & 3;       // 0..3  (N direction)
  const int lx   = lane & 15;
  const int lh   = lane >> 4;     // 0..1

  const int m0 = blockIdx.y * 128;
  const int n0 = blockIdx.x * 128;

  v8f acc[4][2];
#pragma unroll
  for (int mt = 0; mt < 4; ++mt)
#pragma unroll
    for (int nt = 0; nt < 2; ++nt) {
      float bias = b1[n0 + wn * 32 + nt * 16 + lx];
#pragma unroll
      for (int v = 0; v < 8; ++v) acc[mt][nt][v] = bias;
    }

  const int lr = tid >> 1;          // 0..127 : staged row
  const int lc = (tid & 1) * 8;     // 0 or 8 : staged col offset

  for (int kc = 0; kc < 49; ++kc) {
    const int k0 = kc * 16;
    __syncthreads();
    {
      const float* xp = X  + (size_t)(m0 + lr) * D_IN + k0 + lc;
      const float* wp = W1 + (size_t)(n0 + lr) * D_IN + k0 + lc;
#pragma unroll
      for (int j = 0; j < 8; ++j) XS[lr][lc + j] = xp[j];
#pragma unroll
      for (int j = 0; j < 8; ++j) WS[lr][lc + j] = wp[j];
    }
    __syncthreads();

#pragma unroll
    for (int ks = 0; ks < 4; ++ks) {
      const int kk = ks * 4 + lh * 2;     // A: lanes 0-15 K={kk,kk+1}, 16-31 K+2
      v2f a[4], bb[2];
#pragma unroll
      for (int mt = 0; mt < 4; ++mt) {
        int m = wm * 64 + mt * 16 + lx;
        a[mt].x = XS[m][kk];
        a[mt].y = XS[m][kk + 1];
      }
#pragma unroll
      for (int nt = 0; nt < 2; ++nt) {
        int n = wn * 32 + nt * 16 + lx;
        bb[nt].x = WS[n][kk];
        bb[nt].y = WS[n][kk + 1];
      }
#pragma unroll
      for (int mt = 0; mt < 4; ++mt)
#pragma unroll
        for (int nt = 0; nt < 2; ++nt)
          acc[mt][nt] = __builtin_amdgcn_wmma_f32_16x16x4_f32(
              false, a[mt], false, bb[nt], (short)0, acc[mt][nt], false, false);
    }
  }

#pragma unroll
  for (int mt = 0; mt < 4; ++mt)
#pragma unroll
    for (int nt = 0; nt < 2; ++nt) {
      int row = m0 + wm * 64 + mt * 16 + lh * 8;
      int col = n0 + wn * 32 + nt * 16 + lx;
#pragma unroll
      for (int v = 0; v < 8; ++v)
        C[(size_t)(row + v) * D_H + col] = acc[mt][nt][v];
    }
}

// ---------------------------------------------------------------------------
// Kernel 2: 25 leaky-integrate-fire steps. One wave32 owns 2 batch rows.
// mem1 (2 x 2048/32 x f32 = 128 VGPRs) stays in registers across all steps;
// cur1 re-streamed from L2 each step; spk1 @ W2^T done with per-lane FMAs +
// wave32 butterfly reduction; lanes 0..9 carry mem2[b][o] in registers.
// ---------------------------------------------------------------------------
__device__ __forceinline__ float leaky_fire(float& m, float cur) {
  float rst = (m > 1.0f) ? 1.0f : 0.0f;       // heaviside(prev_mem - 1)
  m = fmaf(BETA, m, cur) - rst;               // beta*mem + cur - rst*TH
  return (m > 1.0f) ? 1.0f : 0.0f;            // spike from new mem
}

__global__ __launch_bounds__(256) void snn_steps(
    const float* __restrict__ cur1, const float* __restrict__ W2,
    const float* __restrict__ b2, float* __restrict__ out_spk,
    float* __restrict__ out_mem)
{
  const int lane = threadIdx.x & 31;
  const int wid  = threadIdx.x >> 5;
  const int b0   = (blockIdx.x * 8 + wid) * 2;   // two batch rows per wave

  float4 m1a[16], m1b[16];
#pragma unroll
  for (int i = 0; i < 16; ++i) {
    m1a[i] = make_float4(0.f, 0.f, 0.f, 0.f);
    m1b[i] = make_float4(0.f, 0.f, 0.f, 0.f);
  }
  float m2a = 0.f, m2b = 0.f;
  const float bias2 = (lane < D_OUT) ? b2[lane] : 0.f;

  const float4* c1a = (const float4*)(cur1 + (size_t)b0 * D_H);
  const float4* c1b = (const float4*)(cur1 + (size_t)(b0 + 1) * D_H);
  const float4* W2v = (const float4*)W2;          // [10][512] of float4

  for (int t = 0; t < STEPS; ++t) {
    float acca[D_OUT], accb[D_OUT];
#pragma unroll
    for (int o = 0; o < D_OUT; ++o) { acca[o] = 0.f; accb[o] = 0.f; }

#pragma unroll
    for (int i = 0; i < 16; ++i) {
      const int c = i * 32 + lane;                // float4 chunk index in [0,512)
      float4 ca = c1a[c];
      float4 cb = c1b[c];
      float4 sa, sb;
      sa.x = leaky_fire(m1a[i].x, ca.x);
      sa.y = leaky_fire(m1a[i].y, ca.y);
      sa.z = leaky_fire(m1a[i].z, ca.z);
      sa.w = leaky_fire(m1a[i].w, ca.w);
      sb.x = leaky_fire(m1b[i].x, cb.x);
      sb.y = leaky_fire(m1b[i].y, cb.y);
      sb.z = leaky_fire(m1b[i].z, cb.z);
      sb.w = leaky_fire(m1b[i].w, cb.w);

#pragma unroll
      for (int o = 0; o < D_OUT; ++o) {
        float4 w = W2v[o * (D_H / 4) + c];
        acca[o] = fmaf(sa.x, w.x, fmaf(sa.y, w.y,
                  fmaf(sa.z, w.z, fmaf(sa.w, w.w, acca[o]))));
        accb[o] = fmaf(sb.x, w.x, fmaf(sb.y, w.y,
                  fmaf(sb.z, w.z, fmaf(sb.w, w.w, accb[o]))));
      }
    }

    // butterfly reduction across the 32 lanes of the wave
#pragma unroll
    for (int off = 16; off >= 1; off >>= 1) {
#pragma unroll
      for (int o = 0; o < D_OUT; ++o) {
        acca[o] += __shfl_xor(acca[o], off, 32);
        accb[o] += __shfl_xor(accb[o], off, 32);
      }
    }

    if (lane < D_OUT) {
      const size_t base = (size_t)t * (size_t)(B_SZ * D_OUT) + (size_t)b0 * D_OUT + lane;
      float cur2a = acca[lane] + bias2;
      float rsta  = (m2a > 1.0f) ? 1.0f : 0.0f;
      m2a = fmaf(BETA, m2a, cur2a) - rsta;
      out_spk[base] = (m2a > 1.0f) ? 1.0f : 0.0f;
      out_mem[base] = m2a;

      float cur2b = accb[lane] + bias2;
      float rstb  = (m2b > 1.0f) ? 1.0f : 0.0f;
      m2b = fmaf(BETA, m2b, cur2b) - rstb;
      out_spk[base + D_OUT] = (m2b > 1.0f) ? 1.0f : 0.0f;
      out_mem[base + D_OUT] = m2b;
    }
  }
}

// ---------------------------------------------------------------------------
extern "C" void kernel_launch(void* const* d_in, const int* in_sizes, int n_in,
                              void* d_out, int out_size, void* d_ws, size_t ws_size,
                              hipStream_t stream) {
  (void)in_sizes; (void)n_in; (void)out_size; (void)ws_size;
  const float* x  = (const float*)d_in[0];   // [4096, 784]
  const float* W1 = (const float*)d_in[1];   // [2048, 784]
  const float* b1 = (const float*)d_in[2];   // [2048]
  const float* W2 = (const float*)d_in[3];   // [10, 2048]
  const float* b2 = (const float*)d_in[4];   // [10]

  float* cur1 = (float*)d_ws;                // 4096*2048*4 = 33.5 MB scratch
  float* out_spk = (float*)d_out;                                    // [25,4096,10]
  float* out_mem = out_spk + (size_t)STEPS * B_SZ * D_OUT;           // [25,4096,10]

  dim3 g1(D_H / 128, B_SZ / 128);            // (16, 32)
  gemm1_wmma<<<g1, dim3(256), 0, stream>>>(x, W1, b1, cur1);

  dim3 g2(B_SZ / 16);                        // 256 blocks * 8 waves * 2 rows = 4096
  snn_steps<<<g2, dim3(256), 0, stream>>>(cur1, W2, b2, out_spk, out_mem);
}